// GCNWMMSELayer_SISOAdhoc_81750407512805
// MI455X (gfx1250) — compile-verified
//
#include <hip/hip_runtime.h>
#include <stdint.h>

#define Bdim 128
#define Kdim 512
#define Cdim 4
#define BK (Bdim * Kdim)

#define ROWS_PER_BLK 64          // rows of H per block in pass 1 (4 waves x 16 rows)
#define JCHUNK 32                // K-columns per TDM tile
#define NCHUNK (Kdim / JCHUNK)   // 16
#define TSTRIDE 33               // padded LDS row stride (dwords) -> bank-conflict free

typedef float v2f __attribute__((ext_vector_type(2)));
typedef float v8f __attribute__((ext_vector_type(8)));
typedef unsigned int u32x4 __attribute__((ext_vector_type(4)));
typedef int i32x4 __attribute__((ext_vector_type(4)));
typedef int i32x8 __attribute__((ext_vector_type(8)));

// Low 32 bits of a flat LDS pointer are the LDS byte offset (aperture base is 4GB aligned).
static __device__ __forceinline__ unsigned lds_byte_offset(const void* p) {
  return (unsigned)(uint64_t)(uintptr_t)p;
}

// Issue one TDM 2D tile load (f32 elements) global -> LDS.
// tile_w elements per row, tile_h rows, row stride in elements.
// pad_interval_code=4 -> insert pad_amount(=1 dword) after every 32 dwords,
// giving an LDS row stride of 33 dwords.
static __device__ __forceinline__ void tdm_load_tile_f32(
    unsigned lds_off, const float* gsrc, int tile_w, int tile_h,
    int row_stride_elems, bool pad, int pad_interval_code) {
  uint64_t ga = (uint64_t)(uintptr_t)gsrc;
  u32x4 g0;
  g0[0] = 1u;                                               // count=1 (valid descriptor)
  g0[1] = lds_off;                                          // lds_addr (bytes)
  g0[2] = (unsigned)ga;                                     // global_addr[31:0]
  g0[3] = ((unsigned)(ga >> 32) & 0x01FFFFFFu) | (2u << 30); // global_addr[56:32] | type=2
  unsigned d0 = (2u << 16);                                 // data_size = 4 bytes
  if (pad) d0 |= (1u << 20) | ((unsigned)pad_interval_code << 22); // pad_amount=0 (1 dword)
  i32x8 g1;
  g1[0] = (int)d0;
  g1[1] = 0;                                 // atomic barrier addr = 0, tensor_dim0 lo16 = 0
  g1[2] = 0x10;                              // tensor_dim0 = 1<<20 (no clipping), dim1 lo16 = 0
  g1[3] = 0x10 | (tile_w << 16);             // tensor_dim1 = 1<<20 | tile_dim0
  g1[4] = tile_h & 0xFFFF;                   // tile_dim1 (tile_dim2 = 0 -> 2D)
  g1[5] = row_stride_elems;                  // tensor_dim0_stride lo32
  g1[6] = 0;
  g1[7] = 0;
  i32x4 gz4 = {0, 0, 0, 0};                  // groups 2/3 unused (2D)
  i32x8 gz8 = {0, 0, 0, 0, 0, 0, 0, 0};
  __builtin_amdgcn_tensor_load_to_lds(g0, g1, gz4, gz4, gz8, 0);
}

// ---------------- Pass 1: cov[b,i] = sum_j (hr^2+hi^2) * |v[j]|^2 + noise ----------------
// TDM streams 64x32 tiles of h_re/h_im into LDS (double buffered);
// each wave computes 16 rows via v_wmma_f32_16x16x4_f32 (B = v_pow broadcast over N).
__global__ __launch_bounds__(128) void cov_kernel(
    const float* __restrict__ h_re, const float* __restrict__ h_im,
    const float* __restrict__ v_re, const float* __restrict__ v_im,
    const float* __restrict__ noise, float* __restrict__ cov_out) {
  __shared__ float vp_s[Kdim];
  __shared__ float hre_t[2][ROWS_PER_BLK * TSTRIDE];
  __shared__ float him_t[2][ROWS_PER_BLK * TSTRIDE];

  const int b = blockIdx.x;
  const int i0 = blockIdx.y * ROWS_PER_BLK;
  const int tid = threadIdx.x;
  const int lane = tid & 31;
  const int wid = tid >> 5;     // 0..3
  const int half = lane >> 4;   // 0..1
  const int m = lane & 15;      // 0..15

  // stage |v|^2 for this batch
  for (int j = tid; j < Kdim; j += 128) {
    float vr = v_re[b * Kdim + j], vi = v_im[b * Kdim + j];
    vp_s[j] = vr * vr + vi * vi;
  }

  const float* hre_base = h_re + (size_t)b * Kdim * Kdim + (size_t)i0 * Kdim;
  const float* him_base = h_im + (size_t)b * Kdim * Kdim + (size_t)i0 * Kdim;

  if (wid == 0) {
    tdm_load_tile_f32(lds_byte_offset(&hre_t[0][0]), hre_base, JCHUNK, ROWS_PER_BLK, Kdim, true, 4);
    tdm_load_tile_f32(lds_byte_offset(&him_t[0][0]), him_base, JCHUNK, ROWS_PER_BLK, Kdim, true, 4);
  }

  v8f acc = {0.f, 0.f, 0.f, 0.f, 0.f, 0.f, 0.f, 0.f};

  for (int c = 0; c < NCHUNK; ++c) {
    if (wid == 0) __builtin_amdgcn_s_wait_tensorcnt(0);  // tiles for chunk c landed
    __syncthreads();                                      // publish buffer; prior buffer free
    if (wid == 0 && (c + 1) < NCHUNK) {
      const int nb = (c + 1) & 1;
      tdm_load_tile_f32(lds_byte_offset(&hre_t[nb][0]), hre_base + (c + 1) * JCHUNK,
                        JCHUNK, ROWS_PER_BLK, Kdim, true, 4);
      tdm_load_tile_f32(lds_byte_offset(&him_t[nb][0]), him_base + (c + 1) * JCHUNK,
                        JCHUNK, ROWS_PER_BLK, Kdim, true, 4);
    }
    const int buf = c & 1;
    const int rbase = (wid * 16 + m) * TSTRIDE;
    const int jbase = c * JCHUNK;
#pragma unroll
    for (int kk = 0; kk < JCHUNK / 4; ++kk) {
      const int col = kk * 4 + half * 2;
      // A fragment: lanes 0-15 hold K=0,1 (v0,v1); lanes 16-31 hold K=2,3
      float hr0 = hre_t[buf][rbase + col],     hi0 = him_t[buf][rbase + col];
      float hr1 = hre_t[buf][rbase + col + 1], hi1 = him_t[buf][rbase + col + 1];
      v2f A;
      A.x = hr0 * hr0 + hi0 * hi0;
      A.y = hr1 * hr1 + hi1 * hi1;
      // B fragment: row K replicated across all 16 N lanes -> every D column equal
      v2f Bv;
      Bv.x = vp_s[jbase + col];
      Bv.y = vp_s[jbase + col + 1];
      acc = __builtin_amdgcn_wmma_f32_16x16x4_f32(false, A, false, Bv,
                                                  (short)0, acc, false, false);
    }
  }

  // D layout: vgpr r, lanes 0-15 -> (M=r, N=lane); lanes 16-31 -> (M=r+8, N=lane-16).
  // All N columns are identical; lanes with N==0 write 8 rows each.
  if (m == 0) {
    const int rowg = i0 + wid * 16 + half * 8;
#pragma unroll
    for (int r = 0; r < 8; ++r) {
      cov_out[b * Kdim + rowg + r] = acc[r] + noise[b * Kdim + rowg + r];
    }
  }
}

// ---------------- Pass 2: pointwise u / w / ul / v_tilde ----------------
__global__ __launch_bounds__(256) void pointwise_kernel(
    const float* __restrict__ h_re, const float* __restrict__ h_im,
    const float* __restrict__ v_re, const float* __restrict__ v_im,
    const float* __restrict__ cov,
    float* __restrict__ out_u, float* __restrict__ out_w,
    float* __restrict__ ul_o, float* __restrict__ vtr_o, float* __restrict__ vti_o) {
  const int idx = blockIdx.x * 256 + threadIdx.x;
  const int b = idx >> 9, k = idx & (Kdim - 1);
  const size_t dofs = (size_t)b * Kdim * Kdim + (size_t)k * Kdim + k;
  const float dr = h_re[dofs], di = h_im[dofs];
  const float vr = v_re[idx], vi = v_im[idx];
  const float cv = cov[idx];
  // t = diag * v ; u = t / cov
  const float tr = dr * vr - di * vi, ti = dr * vi + di * vr;
  const float ur = tr / cv, ui = ti / cv;
  // error = 1 - conj(u) * t ; w = 1 / error
  const float er = 1.0f - (ur * tr + ui * ti);
  const float ei = -(ur * ti - ui * tr);
  const float den = er * er + ei * ei;
  const float wr = er / den, wi = -ei / den;
  const float wabs = sqrtf(wr * wr + wi * wi);
  const float uln = (ur * ur + ui * ui) * wabs;
  // v_tilde = conj(diag) * u * w
  const float pr = dr * ur + di * ui, pi = dr * ui - di * ur;
  const float vtr = pr * wr - pi * wi, vti = pr * wi + pi * wr;
  out_u[2 * idx] = ur;  out_u[2 * idx + 1] = ui;
  out_w[2 * idx] = wr;  out_w[2 * idx + 1] = wi;
  ul_o[idx] = uln; vtr_o[idx] = vtr; vti_o[idx] = vti;
}

// ---------------- Pass 3: ul_cov = |Habs2^T @ ul| + fused v-step epilogue ----------------
// Lane = column j -> perfectly coalesced streaming of H; ul broadcast from LDS.
__global__ __launch_bounds__(256) void vstep_kernel(
    const float* __restrict__ h_re, const float* __restrict__ h_im,
    const float* __restrict__ pmax_g,
    const float* __restrict__ ar_re, const float* __restrict__ ar_im,
    const float* __restrict__ dl_re, const float* __restrict__ dl_im,
    const float* __restrict__ rc_re, const float* __restrict__ rc_im,
    const float* __restrict__ ul_i, const float* __restrict__ vtr_i,
    const float* __restrict__ vti_i, float* __restrict__ out_v) {
  __shared__ float ul_s[Kdim];
  const int b = blockIdx.x;
  const int tid = threadIdx.x;
  const int j = blockIdx.y * 256 + tid;
  ul_s[tid] = ul_i[b * Kdim + tid];
  ul_s[tid + 256] = ul_i[b * Kdim + tid + 256];
  __syncthreads();

  const float* hr = h_re + (size_t)b * Kdim * Kdim + j;
  const float* hi = h_im + (size_t)b * Kdim * Kdim + j;
  float a0 = 0.f, a1 = 0.f;
#pragma unroll 8
  for (int i = 0; i < Kdim; i += 2) {
    float x0 = hr[(size_t)i * Kdim],       y0 = hi[(size_t)i * Kdim];
    float x1 = hr[(size_t)(i + 1) * Kdim], y1 = hi[(size_t)(i + 1) * Kdim];
    a0 = fmaf(x0 * x0 + y0 * y0, ul_s[i], a0);
    a1 = fmaf(x1 * x1 + y1 * y1, ul_s[i + 1], a1);
  }
  const float ulcov = fabsf(a0 + a1);

  const int idx = b * Kdim + j;
  const float vtr = vtr_i[idx], vti = vti_i[idx];
  const float pm = pmax_g[idx];
  const float vta = sqrtf(vtr * vtr + vti * vti);
  const float mu = fmaxf(vta / sqrtf(pm) - ulcov, 0.f);
  const float aug = ulcov + mu;
  const float aR = vtr / aug, aI = vti / aug;
  float accR = 0.f, accI = 0.f;
#pragma unroll
  for (int c = 0; c < Cdim; ++c) {
    const float arr = ar_re[c], ari = ar_im[c];
    const float dlr = dl_re[c], dli = dl_im[c];
    float zr = aR * arr - aI * ari + vtr * dlr - vti * dli;
    float zi = aR * ari + aI * arr + vtr * dli + vti * dlr;
    zr = fmaxf(zr, 0.f);  // complex relu
    zi = fmaxf(zi, 0.f);
    const float rr = rc_re[c], ri = rc_im[c];
    accR += zr * rr - zi * ri;
    accI += zr * ri + zi * rr;
  }
  const float cur = fmaxf(pm, accR * accR + accI * accI);
  const float corr = fminf(sqrtf(pm / cur), 1.f);
  out_v[2 * idx] = accR * corr;
  out_v[2 * idx + 1] = accI * corr;
}

extern "C" void kernel_launch(void* const* d_in, const int* in_sizes, int n_in,
                              void* d_out, int out_size, void* d_ws, size_t ws_size,
                              hipStream_t stream) {
  const float* h_re  = (const float*)d_in[0];
  const float* h_im  = (const float*)d_in[1];
  const float* noise = (const float*)d_in[2];
  const float* pmax  = (const float*)d_in[3];
  const float* v_re  = (const float*)d_in[4];
  const float* v_im  = (const float*)d_in[5];
  const float* ar_re = (const float*)d_in[6];
  const float* ar_im = (const float*)d_in[7];
  const float* dl_re = (const float*)d_in[8];
  const float* dl_im = (const float*)d_in[9];
  const float* rc_re = (const float*)d_in[10];
  const float* rc_im = (const float*)d_in[11];

  float* out = (float*)d_out;            // [v_out | u | w], each B*K complex (re,im pairs)
  float* ws  = (float*)d_ws;
  float* cov = ws;                        // B*K
  float* ul  = ws + BK;                   // B*K
  float* vtr = ws + 2 * BK;               // B*K
  float* vti = ws + 3 * BK;               // B*K

  dim3 g1(Bdim, Kdim / ROWS_PER_BLK);     // 128 x 8
  cov_kernel<<<g1, 128, 0, stream>>>(h_re, h_im, v_re, v_im, noise, cov);

  pointwise_kernel<<<BK / 256, 256, 0, stream>>>(h_re, h_im, v_re, v_im, cov,
                                                 out + 2 * BK, out + 4 * BK,
                                                 ul, vtr, vti);

  dim3 g3(Bdim, Kdim / 256);              // 128 x 2
  vstep_kernel<<<g3, 256, 0, stream>>>(h_re, h_im, pmax, ar_re, ar_im, dl_re, dl_im,
                                       rc_re, rc_im, ul, vtr, vti, out);
}